// ModuleDSepconv_51238959841432
// MI455X (gfx1250) — compile-verified
//
#include <hip/hip_runtime.h>
#include <hip/hip_bf16.h>
#include <cstdint>

// DSepConv (deformable separable conv), fixed shapes from the reference:
// inp [2,64,260,260] f32; vertical/horizontal [2,5,256,256];
// offset_x/offset_y/mask [2,25,256,256]; out [2,64,256,256].

#define F_      5
#define TAPS    25
#define TX      32        // pixels per tile (one wave-width of x)
#define NTHR    256       // 8 wave32s
constexpr int Bn   = 2;
constexpr int Cn   = 64;
constexpr int Hn   = 256;
constexpr int Wn   = 256;
constexpr int Hin  = Hn + F_ - 1;   // 260
constexpr int Win  = Wn + F_ - 1;   // 260
constexpr int HW   = Hn * Wn;       // 65536
constexpr int HWin = Hin * Win;     // 67600
constexpr int NROWS = 5 + 5 + TAPS + TAPS + TAPS;  // 85 staged rows

#if defined(__HIP_DEVICE_COMPILE__) && defined(__gfx1250__)
#define ON_GFX1250 1
#else
#define ON_GFX1250 0
#endif

#if ON_GFX1250 && __has_builtin(__builtin_amdgcn_tensor_load_to_lds) && \
    __has_builtin(__builtin_amdgcn_s_wait_tensorcnt)
#define USE_TDM 1
#else
#define USE_TDM 0
#endif

#if ON_GFX1250 && __has_builtin(__builtin_amdgcn_global_load_async_to_lds_b128) && \
    __has_builtin(__builtin_amdgcn_s_wait_asynccnt)
#define USE_ASYNC_LDS 1
#else
#define USE_ASYNC_LDS 0
#endif

#if USE_ASYNC_LDS
typedef int v4i_ __attribute__((ext_vector_type(4)));
typedef __attribute__((address_space(1))) v4i_ as1_v4i;   // global
typedef __attribute__((address_space(3))) v4i_ as3_v4i;   // LDS
#endif

#if USE_TDM
typedef unsigned int u32x4_ __attribute__((ext_vector_type(4)));
typedef int          i32x8_ __attribute__((ext_vector_type(8)));
typedef int          i32x4_ __attribute__((ext_vector_type(4)));

__device__ __forceinline__ unsigned lds_off_of(const void* p) {
    // ptrtoint of a 32-bit LDS (addrspace-3) pointer == LDS byte offset
    return (unsigned)(unsigned long long)
           (__attribute__((address_space(3))) const void*)p;
}

// DMA one 2D tile (nrows x 32 floats, row stride = HW elements) into LDS.
// D# layout per CDNA5 ISA (groups 0..3).
__device__ __forceinline__ void tdm_load_rows(const float* gsrc,
                                              unsigned lds_byte, int nrows) {
    const unsigned long long ga = (unsigned long long)(uintptr_t)gsrc;
    u32x4_ g0;
    g0[0] = 1u;                                       // count=1, user descriptor
    g0[1] = lds_byte;                                 // lds_addr
    g0[2] = (unsigned)ga;                             // global_addr[31:0]
    g0[3] = (unsigned)((ga >> 32) & 0x01ffffffu)      // global_addr[56:32]
            | 0x80000000u;                            // type=2 ("image")
    i32x8_ g1 = {0, 0, 0, 0, 0, 0, 0, 0};
    g1[0] = 0x00020000;                               // data_size=2 (4 bytes)
    g1[2] = 0x4000;                                   // tensor_dim0 = 2^30 (no OOB)
    g1[3] = 0x4000 | (TX << 16);                      // tensor_dim1 = 2^30; tile_dim0 = 32
    g1[4] = nrows;                                    // tile_dim1
    g1[5] = HW;                                       // tensor_dim0_stride = 65536 elems
    i32x4_ z4 = {0, 0, 0, 0};                         // groups 2/3: dims unused (2D tile)
    i32x8_ z8 = {0, 0, 0, 0, 0, 0, 0, 0};             // unused trailing group (6-arg form)
    __builtin_amdgcn_tensor_load_to_lds(g0, g1, z4, z4, z8, 0);
}
#endif

__global__ __launch_bounds__(NTHR) void dsepconv_kernel(
    const float* __restrict__ inp,
    const float* __restrict__ vertical,
    const float* __restrict__ horizontal,
    const float* __restrict__ offx,
    const float* __restrict__ offy,
    const float* __restrict__ mask,
    float* __restrict__ out)
{
    __shared__ __align__(16) float  s_stage[NROWS][TX];  // 10.9 KB
    __shared__ int4   s_off[TAPS][TX];                   // 12.8 KB (byte offsets)
    __shared__ float4 s_cf [TAPS][TX];                   // 12.8 KB

    const int x0 = blockIdx.x * TX;   // tile start in x
    const int y  = blockIdx.y;        // row
    const int b  = blockIdx.z;        // batch
    const int t  = threadIdx.x;

    // ---------------- Phase 0: stage coefficient rows into LDS ------------
    // rows: [0,5)=vertical(i), [5,10)=horizontal(j), [10,35)=mask(k),
    //       [35,60)=offset_x(k), [60,85)=offset_y(k); each row = 32 floats.
#if USE_TDM
    if (t < 32) {   // wave 0 issues 5 TDM descriptors for the whole block
        const size_t rowoff5  = ((size_t)(b * 5)  * Hn + y) * Wn + x0;
        const size_t rowoff25 = ((size_t)(b * 25) * Hn + y) * Wn + x0;
        tdm_load_rows(vertical   + rowoff5,  lds_off_of(&s_stage[0][0]),  5);
        tdm_load_rows(horizontal + rowoff5,  lds_off_of(&s_stage[5][0]),  5);
        tdm_load_rows(mask       + rowoff25, lds_off_of(&s_stage[10][0]), TAPS);
        tdm_load_rows(offx       + rowoff25, lds_off_of(&s_stage[35][0]), TAPS);
        tdm_load_rows(offy       + rowoff25, lds_off_of(&s_stage[60][0]), TAPS);
        __builtin_amdgcn_s_wait_tensorcnt(0);
    }
#else
    for (int item = t; item < NROWS * (TX / 4); item += NTHR) {
        const int r = item >> 3;        // row 0..84
        const int q = item & 7;         // 16B quad within row
        const float* src;
        if      (r < 5)  src = vertical   + (((size_t)(b * 5  + r     ) * Hn + y) * Wn + x0);
        else if (r < 10) src = horizontal + (((size_t)(b * 5  + r - 5 ) * Hn + y) * Wn + x0);
        else if (r < 35) src = mask       + (((size_t)(b * 25 + r - 10) * Hn + y) * Wn + x0);
        else if (r < 60) src = offx       + (((size_t)(b * 25 + r - 35) * Hn + y) * Wn + x0);
        else             src = offy       + (((size_t)(b * 25 + r - 60) * Hn + y) * Wn + x0);
#if USE_ASYNC_LDS
        __builtin_amdgcn_global_load_async_to_lds_b128(
            (as1_v4i*)(src + q * 4),
            (as3_v4i*)&s_stage[r][q * 4],
            0, 0);
#else
        *(float4*)&s_stage[r][q * 4] = *(const float4*)(src + q * 4);
#endif
    }
#if USE_ASYNC_LDS
    __builtin_amdgcn_s_wait_asynccnt(0);
#endif
#endif  // USE_TDM
    __syncthreads();

    // ---- Phase 1: per-(tap,pixel) gather offsets + fused coefficients ----
    // Computed ONCE per pixel, reused by all 64 channels. Offsets stored as
    // BYTE offsets so phase-2 gathers lower to saddr + 32-bit voffset form.
    for (int item = t; item < TAPS * TX; item += NTHR) {
        const int px = item & 31;
        const int k  = item >> 5;            // 0..24
        const int i  = k / F_;
        const int j  = k % F_;

        const float oy = s_stage[60 + k][px];
        const float ox = s_stage[35 + k][px];
        const float w  = s_stage[i][px] * s_stage[5 + j][px] * s_stage[10 + k][px];

        const float py  = (float)(y + i) + oy;
        const float pxf = (float)(x0 + px + j) + ox;
        const float fy = floorf(py);
        const float fx = floorf(pxf);
        const float ay = py - fy;
        const float ax = pxf - fx;
        const int yi = (int)fy;
        const int xi = (int)fx;
        const int y0c = min(max(yi,     0), Hin - 1);
        const int y1c = min(max(yi + 1, 0), Hin - 1);
        const int x0c = min(max(xi,     0), Win - 1);
        const int x1c = min(max(xi + 1, 0), Win - 1);

        const float by = 1.0f - ay, bx = 1.0f - ax;
        s_cf[k][px]  = make_float4(w * by * bx, w * by * ax,
                                   w * ay * bx, w * ay * ax);
        s_off[k][px] = make_int4((y0c * Win + x0c) << 2, (y0c * Win + x1c) << 2,
                                 (y1c * Win + x0c) << 2, (y1c * Win + x1c) << 2);
    }
    __syncthreads();

    // ---------------- Phase 2: gather + accumulate over channels ----------
    const int px = t & 31;           // pixel within tile == lane id
    const int wv = t >> 5;           // wave id 0..7
    // Force the channel base scalar (it is wave-uniform): gathers become
    // global_load_b32 vdst, v_byteoff, s[base] with zero per-load VALU.
    const int cbase = __builtin_amdgcn_readfirstlane(wv * 8);
    const char* __restrict__ pb =
        (const char*)(inp + (size_t)(b * Cn + cbase) * HWin);

    // Warm the caches along the center rows of the gather window.
#pragma unroll
    for (int j = 0; j < 8; ++j) {
        const char* p = pb + (size_t)j * (HWin * 4)
                      + ((size_t)(y + 2) * Win + (x0 + px + 2)) * 4;
        __builtin_prefetch(p, 0, 0);
    }

    float acc[8];
#pragma unroll
    for (int j = 0; j < 8; ++j) acc[j] = 0.0f;

    for (int k = 0; k < TAPS; ++k) {
        const int4   o  = s_off[k][px];
        const float4 cf = s_cf [k][px];
#pragma unroll
        for (int j = 0; j < 8; ++j) {
            const char* __restrict__ p = pb + (size_t)j * (HWin * 4);
            float a = acc[j];
            a = fmaf(cf.x, *(const float*)(p + (unsigned)o.x), a);
            a = fmaf(cf.y, *(const float*)(p + (unsigned)o.y), a);
            a = fmaf(cf.z, *(const float*)(p + (unsigned)o.z), a);
            a = fmaf(cf.w, *(const float*)(p + (unsigned)o.w), a);
            acc[j] = a;
        }
    }

    float* __restrict__ op = out + (((size_t)(b * Cn + cbase) * Hn + y) * Wn + x0 + px);
#pragma unroll
    for (int j = 0; j < 8; ++j) op[(size_t)j * HW] = acc[j];
}

extern "C" void kernel_launch(void* const* d_in, const int* in_sizes, int n_in,
                              void* d_out, int out_size, void* d_ws, size_t ws_size,
                              hipStream_t stream) {
    (void)in_sizes; (void)n_in; (void)out_size; (void)d_ws; (void)ws_size;
    const float* inp        = (const float*)d_in[0];
    const float* vertical   = (const float*)d_in[1];
    const float* horizontal = (const float*)d_in[2];
    const float* offset_x   = (const float*)d_in[3];
    const float* offset_y   = (const float*)d_in[4];
    const float* mask       = (const float*)d_in[5];
    float* out = (float*)d_out;

    dim3 grid(Wn / TX, Hn, Bn);   // 8 x 256 x 2 = 4096 blocks
    dsepconv_kernel<<<grid, NTHR, 0, stream>>>(inp, vertical, horizontal,
                                               offset_x, offset_y, mask, out);
}